// PhongSurfacePy3d_38397007626379
// MI455X (gfx1250) — compile-verified
//
#include <hip/hip_runtime.h>
#include <hip/hip_bf16.h>

typedef float v2f __attribute__((ext_vector_type(2)));
typedef float v8f __attribute__((ext_vector_type(8)));

#define LAMBDA_N    0.01f
#define MAX_DIST    0.1f
#define LR          0.2f
#define LR_DECAY    0.1f
#define ADAGRAD_EPS 1e-10f
#define NORM_EPS    1e-12f

// ---------------------------------------------------------------------------
// Kernel A: face centers packed as (cx, cy, cz, |c|^2); also zero the denom.
// ---------------------------------------------------------------------------
__global__ void centers_kernel(const float* __restrict__ meshV,
                               const int*   __restrict__ meshF,
                               float4* __restrict__ cpack,
                               float* __restrict__ denom,
                               int Nf)
{
    int f = blockIdx.x * blockDim.x + threadIdx.x;
    if (f == 0) *denom = 0.0f;
    if (f >= Nf) return;
    int i0 = meshF[3 * f + 0];
    int i1 = meshF[3 * f + 1];
    int i2 = meshF[3 * f + 2];
    float cx = (meshV[3 * i0 + 0] + meshV[3 * i1 + 0] + meshV[3 * i2 + 0]) / 3.0f;
    float cy = (meshV[3 * i0 + 1] + meshV[3 * i1 + 1] + meshV[3 * i2 + 1]) / 3.0f;
    float cz = (meshV[3 * i0 + 2] + meshV[3 * i1 + 2] + meshV[3 * i2 + 2]) / 3.0f;
    float cc = cx * cx + cy * cy + cz * cz;
    cpack[f] = make_float4(cx, cy, cz, cc);
}

// ---------------------------------------------------------------------------
// Kernel B: WMMA brute-force 1-NN to face centers.
// One wave owns TWO 16-row query tiles (32 rows) and loops over all Nf/16
// face tiles; each B tile (single global_load_b64 per lane) feeds TWO
// independent v_wmma_f32_16x16x4_f32 issues. Software-pipelined double
// buffer on the B operand overlaps the next load with the current WMMAs.
//
// A (16x4 f32): lane<16 holds (K0,K1)=( -2qx, -2qy ) for M=lane;
//               lane>=16 holds (K2,K3)=( -2qz, 1.0 ) for M=lane-16.
// B (4x16 f32): lane<16 holds (K0,K1)=( cx, cy ) for N=lane;
//               lane>=16 holds (K2,K3)=( cz, |c|^2 ) for N=lane-16.
// D (16x16 f32): slot r, lanes 0-15 -> M=r, N=lane;
//                        lanes 16-31 -> M=r+8, N=lane-16.
// ---------------------------------------------------------------------------
__global__ void __launch_bounds__(256)
knn_wmma_kernel(const float*  __restrict__ qV,
                const float*  __restrict__ cpackf,   // Nf * 4 floats
                int*   __restrict__ fidx_ws,
                float* __restrict__ mask_ws,
                float* __restrict__ denom_ws,
                float* __restrict__ out_fidx,
                float* __restrict__ out_outlier,
                int Np, int Nf)
{
    const int lane = threadIdx.x & 31;
    const int wave = blockIdx.x * (blockDim.x >> 5) + (threadIdx.x >> 5);
    const int row0 = wave * 32;
    if (row0 >= Np) return;                 // full waves only (Np multiple of 32)

    const int ln15 = lane & 15;
    const int hi   = (lane >> 4);           // 0 for lanes 0-15, 1 for 16-31
    const int sel  = hi * 2;                // float offset within packed center

    // A operands for the two query tiles
    const int m0 = row0 + ln15;
    const int m1 = row0 + 16 + ln15;
    v2f a0, a1;
    {
        float q0x = qV[3 * m0 + 0], q0y = qV[3 * m0 + 1], q0z = qV[3 * m0 + 2];
        float q1x = qV[3 * m1 + 0], q1y = qV[3 * m1 + 1], q1z = qV[3 * m1 + 2];
        if (!hi) { a0.x = -2.0f * q0x; a0.y = -2.0f * q0y;
                   a1.x = -2.0f * q1x; a1.y = -2.0f * q1y; }
        else     { a0.x = -2.0f * q0z; a0.y = 1.0f;
                   a1.x = -2.0f * q1z; a1.y = 1.0f; }
    }

    float bv0[8], bv1[8];
    int   bt0[8], bt1[8];
#pragma unroll
    for (int r = 0; r < 8; ++r) {
        bv0[r] = 3.4e38f; bt0[r] = 0;
        bv1[r] = 3.4e38f; bt1[r] = 0;
    }

    const int ntiles = Nf >> 4;
    // per-lane B pointer: lane reads 2 consecutive floats of cpack[t*16 + ln15]
    const float* bptr = cpackf + 4 * ln15 + sel;

    v2f bcur = *(const v2f*)(bptr);         // tile 0
    for (int t = 0; t < ntiles; ++t) {
        int tn = t + 1 < ntiles ? t + 1 : t;       // clamped, branch-free
        v2f bnext = *(const v2f*)(bptr + (size_t)tn * 64);
        if (t + 8 < ntiles)
            __builtin_prefetch(bptr + (size_t)(t + 8) * 64, 0, 1);

        v8f acc0 = {}, acc1 = {};
        acc0 = __builtin_amdgcn_wmma_f32_16x16x4_f32(
            false, a0, false, bcur, (short)0, acc0, false, false);
        acc1 = __builtin_amdgcn_wmma_f32_16x16x4_f32(
            false, a1, false, bcur, (short)0, acc1, false, false);

#pragma unroll
        for (int r = 0; r < 8; ++r) {
            float v0 = acc0[r];
            if (v0 < bv0[r]) { bv0[r] = v0; bt0[r] = t; }
            float v1 = acc1[r];
            if (v1 < bv1[r]) { bv1[r] = v1; bt1[r] = t; }
        }
        bcur = bnext;
    }

    int bi0[8], bi1[8];
#pragma unroll
    for (int r = 0; r < 8; ++r) {
        bi0[r] = bt0[r] * 16 + ln15;
        bi1[r] = bt1[r] * 16 + ln15;
    }

    // 16-lane xor butterfly (stays within each half of the wave32)
#pragma unroll
    for (int msk = 1; msk < 16; msk <<= 1) {
#pragma unroll
        for (int r = 0; r < 8; ++r) {
            float ov = __shfl_xor(bv0[r], msk, 32);
            int   oi = __shfl_xor(bi0[r], msk, 32);
            if (ov < bv0[r] || (ov == bv0[r] && oi < bi0[r])) { bv0[r] = ov; bi0[r] = oi; }
            ov = __shfl_xor(bv1[r], msk, 32);
            oi = __shfl_xor(bi1[r], msk, 32);
            if (ov < bv1[r] || (ov == bv1[r] && oi < bi1[r])) { bv1[r] = ov; bi1[r] = oi; }
        }
    }

    if (lane == 0 || lane == 16) {
        float msum = 0.0f;
#pragma unroll
        for (int g = 0; g < 2; ++g) {       // g=0: tile0 slots, g=1: tile1 slots
            const int base = row0 + g * 16 + hi * 8;
#pragma unroll
            for (int r = 0; r < 8; ++r) {
                int row = base + r;
                float rx = qV[3 * row + 0];
                float ry = qV[3 * row + 1];
                float rz = qV[3 * row + 2];
                float qq = rx * rx + ry * ry + rz * rz;
                float bv = g ? bv1[r] : bv0[r];
                int   bi = g ? bi1[r] : bi0[r];
                float d2 = bv + qq;
                bool outlier = d2 > MAX_DIST;
                float mk = outlier ? 0.0f : 1.0f;
                fidx_ws[row]     = bi;
                mask_ws[row]     = mk;
                out_fidx[row]    = (float)bi;
                out_outlier[row] = outlier ? 1.0f : 0.0f;
                msum += mk;
            }
        }
        atomicAdd(denom_ws, msum);   // sums of 0/1 floats: exact, order-independent
    }
}

// ---------------------------------------------------------------------------
// Kernel C: per-point Adagrad refinement (2 outer x 50 inner), analytic grad.
// ---------------------------------------------------------------------------
__global__ void phong_opt_kernel(const float* __restrict__ qV,
                                 const float* __restrict__ qN,
                                 const float* __restrict__ meshV,
                                 const int*   __restrict__ meshF,
                                 const float* __restrict__ meshN,
                                 const int*   __restrict__ fidx_ws,
                                 const float* __restrict__ mask_ws,
                                 const float* __restrict__ denom_ws,
                                 float* __restrict__ out_vw,
                                 int Np)
{
    int n = blockIdx.x * blockDim.x + threadIdx.x;
    if (n >= Np) return;

    const float denom = fmaxf(denom_ws[0], 1.0f);
    const float mk    = mask_ws[n];
    const float scale = mk / denom;
    const int f = fidx_ws[n];

    const int i0 = meshF[3 * f + 0];
    const int i1 = meshF[3 * f + 1];
    const int i2 = meshF[3 * f + 2];

    float V0x = meshV[3*i0+0], V0y = meshV[3*i0+1], V0z = meshV[3*i0+2];
    float V1x = meshV[3*i1+0], V1y = meshV[3*i1+1], V1z = meshV[3*i1+2];
    float V2x = meshV[3*i2+0], V2y = meshV[3*i2+1], V2z = meshV[3*i2+2];
    float N0x = meshN[3*i0+0], N0y = meshN[3*i0+1], N0z = meshN[3*i0+2];
    float N1x = meshN[3*i1+0], N1y = meshN[3*i1+1], N1z = meshN[3*i1+2];
    float N2x = meshN[3*i2+0], N2y = meshN[3*i2+1], N2z = meshN[3*i2+2];

    const float qx = qV[3*n+0], qy = qV[3*n+1], qz = qV[3*n+2];
    const float qnx = qN[3*n+0], qny = qN[3*n+1], qnz = qN[3*n+2];

    // dP/dv = V0-V2, dP/dw = V1-V2 ; same for normals
    const float E0x = V0x - V2x, E0y = V0y - V2y, E0z = V0z - V2z;
    const float E1x = V1x - V2x, E1y = V1y - V2y, E1z = V1z - V2z;
    const float F0x = N0x - N2x, F0y = N0y - N2y, F0z = N0z - N2z;
    const float F1x = N1x - N2x, F1y = N1y - N2y, F1z = N1z - N2z;

    float vb = 1.0f / 3.0f, wb = 1.0f / 3.0f;
    float alpha = 1.0f;

    for (int outer = 0; outer < 2; ++outer) {
        float d0 = 0.0f, d1 = 0.0f;   // delta
        float s0 = 0.0f, s1 = 0.0f;   // adagrad sum of squares
        for (int i = 0; i < 50; ++i) {
            float v = vb + d0, w = wb + d1, u = 1.0f - v - w;
            // position term
            float Px = v * V0x + w * V1x + u * V2x;
            float Py = v * V0y + w * V1y + u * V2y;
            float Pz = v * V0z + w * V1z + u * V2z;
            float ex = Px - qx, ey = Py - qy, ez = Pz - qz;
            float lv = sqrtf(ex * ex + ey * ey + ez * ez);
            float ilv = 1.0f / lv;
            float gPx = ex * ilv, gPy = ey * ilv, gPz = ez * ilv;
            // normal term
            float Nx = v * N0x + w * N1x + u * N2x;
            float Ny = v * N0y + w * N1y + u * N2y;
            float Nz = v * N0z + w * N1z + u * N2z;
            float r  = sqrtf(Nx * Nx + Ny * Ny + Nz * Nz);
            float rm = fmaxf(r, NORM_EPS);
            float irm = 1.0f / rm;
            float nx = Nx * irm, ny = Ny * irm, nz = Nz * irm;
            float fx = nx - qnx, fy = ny - qny, fz = nz - qnz;
            float ln = sqrtf(fx * fx + fy * fy + fz * fz);
            float iln = 1.0f / ln;
            float gnx = fx * iln, gny = fy * iln, gnz = fz * iln;
            float ndg = nx * gnx + ny * gny + nz * gnz;
            float gNx = (gnx - nx * ndg) * irm;
            float gNy = (gny - ny * ndg) * irm;
            float gNz = (gnz - nz * ndg) * irm;

            float gv = gPx * E0x + gPy * E0y + gPz * E0z
                     + LAMBDA_N * (gNx * F0x + gNy * F0y + gNz * F0z);
            float gw = gPx * E1x + gPy * E1y + gPz * E1z
                     + LAMBDA_N * (gNx * F1x + gNy * F1y + gNz * F1z);
            gv *= scale; gw *= scale;

            s0 += gv * gv; s1 += gw * gw;
            float clr = LR / (1.0f + (float)i * LR_DECAY);
            d0 -= clr * gv / (sqrtf(s0) + ADAGRAD_EPS);
            d1 -= clr * gw / (sqrtf(s1) + ADAGRAD_EPS);
        }
        vb += d0 * alpha; wb += d1 * alpha;
        alpha *= 0.5f;
    }

    bool valid = mk > 0.5f;
    out_vw[2 * n + 0] = valid ? vb : (1.0f / 3.0f);
    out_vw[2 * n + 1] = valid ? wb : (1.0f / 3.0f);
}

// ---------------------------------------------------------------------------
// Launch
// ---------------------------------------------------------------------------
extern "C" void kernel_launch(void* const* d_in, const int* in_sizes, int n_in,
                              void* d_out, int out_size, void* d_ws, size_t ws_size,
                              hipStream_t stream)
{
    const float* qV    = (const float*)d_in[0];   // [B,Np,3]
    const float* qN    = (const float*)d_in[1];   // [B,Np,3]
    const float* meshV = (const float*)d_in[2];   // [Nv,3]
    const int*   meshF = (const int*)  d_in[3];   // [Nf,3]
    const float* meshN = (const float*)d_in[4];   // [Nv,3]

    const int Np = in_sizes[0] / 3;
    const int Nf = in_sizes[3] / 3;

    // workspace layout
    char* ws = (char*)d_ws;
    float4* cpack   = (float4*)(ws);                         // Nf * 16 bytes
    int*    fidx_ws = (int*)   (ws + (size_t)Nf * 16);       // Np * 4
    float*  mask_ws = (float*) (ws + (size_t)Nf * 16 + (size_t)Np * 4);
    float*  denom   = (float*) (ws + (size_t)Nf * 16 + (size_t)Np * 8);

    // output layout (return order: fidx, vw, outlier_mask)
    float* out_fidx    = (float*)d_out;           // Np
    float* out_vw      = out_fidx + Np;           // Np*2
    float* out_outlier = out_vw + 2 * Np;         // Np

    // A: centers + |c|^2, zero denom
    {
        int threads = 256, blocks = (Nf + threads - 1) / threads;
        centers_kernel<<<blocks, threads, 0, stream>>>(meshV, meshF, cpack, denom, Nf);
    }

    // B: WMMA KNN; one wave per 32 query rows (two 16x16 output tiles / B load)
    {
        int waves = Np / 32;
        int threads = 256;                         // 8 waves per block
        int blocks = (waves * 32 + threads - 1) / threads;
        knn_wmma_kernel<<<blocks, threads, 0, stream>>>(
            qV, (const float*)cpack, fidx_ws, mask_ws, denom,
            out_fidx, out_outlier, Np, Nf);
    }

    // C: per-point Adagrad refinement
    {
        int threads = 256, blocks = (Np + threads - 1) / threads;
        phong_opt_kernel<<<blocks, threads, 0, stream>>>(
            qV, qN, meshV, meshF, meshN, fidx_ws, mask_ws, denom, out_vw, Np);
    }
}